// GNNEncoder_80324478369999
// MI455X (gfx1250) — compile-verified
//
#include <hip/hip_runtime.h>

// 2-layer GCN for MI455X (gfx1250, wave32).
//   Layer: out = D^-1/2 A D^-1/2 (x @ W) + b  (A includes self loops)
// GEMM (3.3 GFLOP total) done with full-precision V_WMMA_F32_16X16X4_F32.
// Edge aggregation (the real cost, ~870MB/layer of irregular traffic) is
// L2-resident on MI455X (3 x 25.6MB feature buffers << 192MB L2) and uses
// hardware f32 global atomics with coalesced float4 gathers.

#define NDIM 128  // feature width (fixed by reference)

typedef __attribute__((ext_vector_type(2))) float v2f;
typedef __attribute__((ext_vector_type(8))) float v8f;

// ---------------- degree / normalization ----------------

__global__ void gcn_deg_init(float* __restrict__ deg, int n) {
    int i = blockIdx.x * blockDim.x + threadIdx.x;
    if (i < n) deg[i] = 1.0f;  // self-loop contributes 1 to every node
}

__global__ void gcn_deg_accum(const int* __restrict__ dst, float* __restrict__ deg, int E) {
    int e = blockIdx.x * blockDim.x + threadIdx.x;
    if (e < E) unsafeAtomicAdd(&deg[dst[e]], 1.0f);  // global_atomic_add_f32
}

__global__ void gcn_dinv(float* __restrict__ deg, int n) {
    int i = blockIdx.x * blockDim.x + threadIdx.x;
    if (i < n) deg[i] = rsqrtf(deg[i]);  // deg >= 1 always
}

// ---------------- GEMM: H = (RELU? max(A,0):A) @ W  via f32 WMMA ----------------
// One wave per 16-row strip of A; computes all 8 16x16 column tiles
// (8 x v8f accumulators). n must be a multiple of 16 (50000 = 3125*16).

template <bool RELU>
__global__ __launch_bounds__(256) void gcn_gemm_wmma(const float* __restrict__ A,
                                                     const float* __restrict__ W,
                                                     float* __restrict__ H,
                                                     int nStrips) {
    const int wave  = threadIdx.x >> 5;
    const int strip = blockIdx.x * 8 + wave;
    if (strip >= nStrips) return;  // wave-uniform: EXEC stays all-1s for WMMA
    const int lane = threadIdx.x & 31;
    const int half = lane >> 4;   // 0: K=k,k+1 / M rows 0..7 ; 1: K=k+2,k+3 / M rows 8..15
    const int l16  = lane & 15;
    const int row0 = strip * 16;

    const float* __restrict__ arow = A + (size_t)(row0 + l16) * NDIM;

    v8f acc[8] = {};

    for (int k = 0; k < NDIM; k += 4) {
        const int kk = k + 2 * half;
        // A fragment: 16x4 f32, one row per lane-group, 2 K-values per lane
        float a0 = arow[kk];
        float a1 = arow[kk + 1];
        if (RELU) { a0 = fmaxf(a0, 0.0f); a1 = fmaxf(a1, 0.0f); }
        v2f a; a.x = a0; a.y = a1;
#pragma unroll
        for (int t = 0; t < 8; ++t) {
            // B fragment: 4x16 f32 tile of W (row-major [K][N]); W stays hot in cache
            const int col = t * 16 + l16;
            v2f b;
            b.x = W[(size_t)kk * NDIM + col];
            b.y = W[(size_t)(kk + 1) * NDIM + col];
            acc[t] = __builtin_amdgcn_wmma_f32_16x16x4_f32(
                false, a, false, b, (short)0, acc[t], false, false);
        }
    }

    // C/D layout: VGPR r -> M = r + 8*half, N = l16
#pragma unroll
    for (int t = 0; t < 8; ++t) {
        const int col = t * 16 + l16;
#pragma unroll
        for (int r = 0; r < 8; ++r) {
            const int row = row0 + r + 8 * half;
            H[(size_t)row * NDIM + col] = acc[t][r];
        }
    }
}

// ---------------- node init: out = h * dinv^2 (self loop) + bias ----------------

__global__ void gcn_node_init(const float* __restrict__ h, const float* __restrict__ dinv,
                              const float* __restrict__ bias, float* __restrict__ out,
                              int n) {
    int idx = blockIdx.x * blockDim.x + threadIdx.x;
    if (idx < n * NDIM) {
        int i = idx >> 7;       // node
        int c = idx & (NDIM - 1);
        float di = dinv[i];
        out[idx] = h[idx] * di * di + bias[c];
    }
}

// ---------------- edge aggregation: out[dst] += h[src] * dinv[src]*dinv[dst] ----------------
// One wave32 per edge; lane loads float4 (coalesced 512B/wave) and issues 4
// hardware f32 atomics. Both h and out live in the 192MB L2.

__global__ __launch_bounds__(256) void gcn_edge_agg(const int* __restrict__ src,
                                                    const int* __restrict__ dst,
                                                    const float* __restrict__ dinv,
                                                    const float* __restrict__ h,
                                                    float* __restrict__ out, int E) {
    const int e = blockIdx.x * 8 + (threadIdx.x >> 5);
    if (e >= E) return;
    const int lane = threadIdx.x & 31;
    const int s = src[e];
    const int d = dst[e];
    const float w = dinv[s] * dinv[d];
    const float4 v = ((const float4*)(h + (size_t)s * NDIM))[lane];
    float* op = out + (size_t)d * NDIM + lane * 4;
    unsafeAtomicAdd(op + 0, v.x * w);
    unsafeAtomicAdd(op + 1, v.y * w);
    unsafeAtomicAdd(op + 2, v.z * w);
    unsafeAtomicAdd(op + 3, v.w * w);
}

// ---------------- launch ----------------

extern "C" void kernel_launch(void* const* d_in, const int* in_sizes, int n_in,
                              void* d_out, int out_size, void* d_ws, size_t ws_size,
                              hipStream_t stream) {
    const float* x  = (const float*)d_in[0];
    const int*   ei = (const int*)d_in[1];   // [2, E]: row0=src, row1=dst
    const float* W1 = (const float*)d_in[2];
    const float* b1 = (const float*)d_in[3];
    const float* W2 = (const float*)d_in[4];
    const float* b2 = (const float*)d_in[5];
    float* out = (float*)d_out;

    const int n = in_sizes[0] / NDIM;  // 50000
    const int E = in_sizes[1] / 2;     // 800000
    const int* src = ei;
    const int* dst = ei + E;

    // workspace: deg/dinv [n] | h1 [n*128] | a1 [n*128] | h2 [n*128]
    float* deg = (float*)d_ws;
    float* h1  = deg + n;
    float* a1  = h1 + (size_t)n * NDIM;
    float* h2  = a1 + (size_t)n * NDIM;

    const int nStrips = (n + 15) / 16;
    const dim3 blk(256);
    const int gN  = (n + 255) / 256;
    const int gNC = (int)(((size_t)n * NDIM + 255) / 256);
    const int gE  = (E + 255) / 256;
    const int gEw = (E + 7) / 8;           // one wave per edge, 8 waves/block
    const int gS  = (nStrips + 7) / 8;     // one wave per 16-row strip

    // normalization (shared by both layers)
    gcn_deg_init <<<gN, blk, 0, stream>>>(deg, n);
    gcn_deg_accum<<<gE, blk, 0, stream>>>(dst, deg, E);
    gcn_dinv     <<<gN, blk, 0, stream>>>(deg, n);

    // layer 1
    gcn_gemm_wmma<false><<<gS, blk, 0, stream>>>(x, W1, h1, nStrips);
    gcn_node_init<<<gNC, blk, 0, stream>>>(h1, deg, b1, a1, n);
    gcn_edge_agg <<<gEw, blk, 0, stream>>>(src, dst, deg, h1, a1, E);

    // layer 2 (ReLU of a1 fused into the GEMM's A-fragment loads)
    gcn_gemm_wmma<true><<<gS, blk, 0, stream>>>(a1, W2, h2, nStrips);
    gcn_node_init<<<gNC, blk, 0, stream>>>(h2, deg, b2, out, n);
    gcn_edge_agg <<<gEw, blk, 0, stream>>>(src, dst, deg, h2, out, E);
}